// EdgeSamplingGumbel_27118423507706
// MI455X (gfx1250) — compile-verified
//
#include <hip/hip_runtime.h>
#include <math.h>

// ---------------------------------------------------------------------------
// Fused Poincare-distance + Gumbel-softmax top-k edge sampler for MI455X.
//
//  N=10000, D=128, K=5.
//  - Gram matrix via V_WMMA_F32_16X16X4_F32 (exact fp32 tensor path).
//  - Single pass over q (400 MB, the only O(N^2) HBM traffic, ~17us floor
//    at 23.3 TB/s). dist / y_soft are never materialized.
//  - Online softmax + running top-5 per row, merged across lane halves.
// ---------------------------------------------------------------------------

#define N_PTS   10000
#define DIM     128
#define TOPK    5
#define TILE    16
#define NTILES  (N_PTS / TILE)          /* 625 (N divisible by 16) */
#define EPS_BALL  1e-2f
#define EPS_ACOSH 1e-6f
#define NEG_BIG  (-3.402823466e38f)

typedef float v2f __attribute__((ext_vector_type(2)));
typedef float v4f __attribute__((ext_vector_type(4)));
typedef float v8f __attribute__((ext_vector_type(8)));

// ---------------- projection: x -> xp (shrunk Poincare ball), xn = |xp|^2 ---
__global__ __launch_bounds__(128) void proj_kernel(const float* __restrict__ x,
                                                   float* __restrict__ xp,
                                                   float* __restrict__ xn) {
    const int wid  = threadIdx.x >> 5;
    const int lane = threadIdx.x & 31;
    const int row  = blockIdx.x * 4 + wid;          // one wave per row
    if (row >= N_PTS) return;
    const float* xr = x + (size_t)row * DIM + lane * 4;
    v4f v = *(const v4f*)xr;
    float s = v.x * v.x + v.y * v.y + v.z * v.z + v.w * v.w;
    #pragma unroll
    for (int off = 16; off > 0; off >>= 1) s += __shfl_xor(s, off, 32);
    const float norm  = sqrtf(s);
    const float scale = 1.0f / ((fmaxf(norm - 1.0f, 0.0f) + 1.0f) * (1.0f + EPS_BALL));
    v4f o = { v.x * scale, v.y * scale, v.z * scale, v.w * scale };
    *(v4f*)(xp + (size_t)row * DIM + lane * 4) = o;
    if (lane == 0) xn[row] = s * scale * scale;
}

// ---------------- copy input x to output slot 0 (reference returns raw x) ---
__global__ __launch_bounds__(256) void copy_kernel(const float* __restrict__ src,
                                                   float* __restrict__ dst) {
    const size_t i = (size_t)blockIdx.x * 256 + threadIdx.x;
    dst[i] = src[i];
}

// top-5 insertion ladder kept entirely in scalar registers (no arrays)
#define TOP5_INSERT(sv, sc)                                                    \
    if ((sv) > t4v) {                                                          \
        if ((sv) > t2v) {                                                      \
            if ((sv) > t1v) {                                                  \
                if ((sv) > t0v) {                                              \
                    t4v=t3v; t4i=t3i; t3v=t2v; t3i=t2i; t2v=t1v; t2i=t1i;      \
                    t1v=t0v; t1i=t0i; t0v=(sv); t0i=(sc);                      \
                } else {                                                       \
                    t4v=t3v; t4i=t3i; t3v=t2v; t3i=t2i; t2v=t1v; t2i=t1i;      \
                    t1v=(sv); t1i=(sc);                                        \
                }                                                              \
            } else {                                                           \
                t4v=t3v; t4i=t3i; t3v=t2v; t3i=t2i; t2v=(sv); t2i=(sc);        \
            }                                                                  \
        } else {                                                               \
            if ((sv) > t3v) { t4v=t3v; t4i=t3i; t3v=(sv); t3i=(sc); }          \
            else            { t4v=(sv); t4i=(sc); }                            \
        }                                                                      \
    }

// ---------------- main fused kernel: one wave per 16-row tile ---------------
__global__ __launch_bounds__(128) void edge_kernel(const float* __restrict__ xp,
                                                   const float* __restrict__ xn,
                                                   const float* __restrict__ q,
                                                   const float* __restrict__ temp_p,
                                                   float* __restrict__ out_rows,
                                                   float* __restrict__ out_cols,
                                                   float* __restrict__ out_w) {
    __shared__ float lds[4][TILE * 17];             // wave-private, padded
    const int wid  = threadIdx.x >> 5;
    const int lane = threadIdx.x & 31;
    const int tile = blockIdx.x * 4 + wid;
    if (tile >= NTILES) return;                     // no barriers used below
    const int r0   = tile * TILE;
    const int lrow = lane & 15;                     // M (A) / N (B,C) index
    const int half = lane >> 4;                     // 0 or 1

    const float t     = fminf(fmaxf(temp_p[0], 0.0f), 5.0f);
    const float net   = -__expf(t);                 // logits = dist * net
    const float inv_t = 1.0f / t;

    // ---- preload A operand: 16x128 fp32, ISA 16x4 layout per wmma step ----
    // lane slot j of step k0 holds xp[r0 + lrow][4*k0 + 2*half + j]
    v2f a[32];
    const float* ap = xp + (size_t)(r0 + lrow) * DIM + 2 * half;
    #pragma unroll
    for (int k0 = 0; k0 < 32; ++k0) a[k0] = *(const v2f*)(ap + 4 * k0);

    float xnr[8];
    #pragma unroll
    for (int v = 0; v < 8; ++v) xnr[v] = xn[r0 + v + 8 * half];   // row norms

    // online softmax + top-5 state (lane owns row lrow, cols half*8..half*8+7)
    float mx = NEG_BIG, sum = 0.0f;
    float t0v = NEG_BIG, t1v = NEG_BIG, t2v = NEG_BIG, t3v = NEG_BIG, t4v = NEG_BIG;
    int   t0i = 0, t1i = 0, t2i = 0, t3i = 0, t4i = 0;

    for (int ct = 0; ct < NTILES; ++ct) {
        const int c0 = ct * TILE;
        // B operand: 4x16 fp32 per step; lane slot j = xp[c0+lrow][4*k0+2*half+j]
        const float* bp  = xp + (size_t)(c0 + lrow) * DIM + 2 * half;
        const float  xnc = xn[c0 + lrow];

        v8f acc = {};
        #pragma unroll
        for (int k0 = 0; k0 < 32; ++k0) {
            v2f b = *(const v2f*)(bp + 4 * k0);
            acc = __builtin_amdgcn_wmma_f32_16x16x4_f32(
                false, a[k0], false, b, (short)0, acc, false, false);
        }

        // ---- fused elementwise transform; C slot v is (m = v+8*half, n = lrow)
        const float omc   = 1.0f - xnc;
        const size_t qbase = (size_t)(r0 + 8 * half) * N_PTS + (size_t)(c0 + lrow);
        #pragma unroll
        for (int v = 0; v < 8; ++v) {
            const float g  = acc[v];
            const float xr = xnr[v];
            float pq = fmaxf(xr + xnc - 2.0f * g, 0.0f);
            float z  = 1.0f + EPS_ACOSH + 2.0f * pq / (omc * (1.0f - xr));
            float ac = __logf(z + sqrtf(z * z - 1.0f));     // arccosh(z)
            float d2 = ac * ac;
            float qv  = q[qbase + (size_t)v * N_PTS];       // the 400MB stream
            float gum = -__logf(-__logf(qv));               // Gumbel noise
            float s   = (d2 * net + gum) * inv_t;
            lds[wid][(v + 8 * half) * 17 + lrow] = s;
        }
        __asm volatile("s_wait_dscnt 0x0" ::: "memory");    // wave-local LDS sync

        // ---- scan: lane owns row lrow, 8 columns of this tile ----
        #pragma unroll
        for (int n = 0; n < 8; ++n) {
            const int   cc  = half * 8 + n;
            const float s   = lds[wid][lrow * 17 + cc];
            const int   col = c0 + cc;
            if (s > mx) { sum = sum * __expf(mx - s) + 1.0f; mx = s; }
            else        { sum += __expf(s - mx); }
            TOP5_INSERT(s, col)
        }
    }

    // ---- merge the two half-lanes owning the same row ----
    const float pmx  = __shfl_xor(mx, 16, 32);
    const float psum = __shfl_xor(sum, 16, 32);
    const float M    = fmaxf(mx, pmx);
    const float tot  = sum * __expf(mx - M) + psum * __expf(pmx - M);
    {
        float pv; int pi;
        pv = __shfl_xor(t0v, 16, 32); pi = __shfl_xor(t0i, 16, 32); TOP5_INSERT(pv, pi)
        pv = __shfl_xor(t1v, 16, 32); pi = __shfl_xor(t1i, 16, 32); TOP5_INSERT(pv, pi)
        pv = __shfl_xor(t2v, 16, 32); pi = __shfl_xor(t2i, 16, 32); TOP5_INSERT(pv, pi)
        pv = __shfl_xor(t3v, 16, 32); pi = __shfl_xor(t3i, 16, 32); TOP5_INSERT(pv, pi)
        pv = __shfl_xor(t4v, 16, 32); pi = __shfl_xor(t4i, 16, 32); TOP5_INSERT(pv, pi)
    }

    if (half == 0) {
        const int   row  = r0 + lrow;
        const float itot = 1.0f / tot;
        const int   base = row * TOPK;
        out_rows[base + 0] = (float)row;  out_cols[base + 0] = (float)t0i;
        out_rows[base + 1] = (float)row;  out_cols[base + 1] = (float)t1i;
        out_rows[base + 2] = (float)row;  out_cols[base + 2] = (float)t2i;
        out_rows[base + 3] = (float)row;  out_cols[base + 3] = (float)t3i;
        out_rows[base + 4] = (float)row;  out_cols[base + 4] = (float)t4i;
        out_w[base + 0] = __expf(t0v - M) * itot;
        out_w[base + 1] = __expf(t1v - M) * itot;
        out_w[base + 2] = __expf(t2v - M) * itot;
        out_w[base + 3] = __expf(t3v - M) * itot;
        out_w[base + 4] = __expf(t4v - M) * itot;
    }
}

extern "C" void kernel_launch(void* const* d_in, const int* in_sizes, int n_in,
                              void* d_out, int out_size, void* d_ws, size_t ws_size,
                              hipStream_t stream) {
    const float* x    = (const float*)d_in[0];   // [N, D] f32
    const float* q    = (const float*)d_in[1];   // [N, N] f32
    const float* temp = (const float*)d_in[2];   // scalar f32
    float* out = (float*)d_out;

    // workspace: projected xp [N*D] then norms xn [N]  (~5.2 MB)
    float* xp = (float*)d_ws;
    float* xn = xp + (size_t)N_PTS * DIM;

    // out layout: x [N*D] | edge rows [N*K] | edge cols [N*K] | weights [N*K]
    float* out_rows = out + (size_t)N_PTS * DIM;
    float* out_cols = out_rows + (size_t)N_PTS * TOPK;
    float* out_w    = out_cols + (size_t)N_PTS * TOPK;

    proj_kernel<<<N_PTS / 4, 128, 0, stream>>>(x, xp, xn);
    copy_kernel<<<(N_PTS * DIM) / 256, 256, 0, stream>>>(x, out);
    edge_kernel<<<(NTILES + 3) / 4, 128, 0, stream>>>(xp, xn, q, temp,
                                                      out_rows, out_cols, out_w);
}